// CrossCompressUnit_3607772529077
// MI455X (gfx1250) — compile-verified
//
#include <hip/hip_runtime.h>

// CrossCompressUnit for MI455X (gfx1250, wave32).
// Memory-bound (~1 flop/byte): floor = 256 MiB / 23.3 TB/s ~= 11 us.
// Phase 1: per-16-row-tile dot-product GEMM [16xD]x[Dx4] via V_WMMA_F32_16X16X4_F32.
// Phase 2: streaming broadcast-FMA with b128 loads + non-temporal b128 stores.

typedef __attribute__((ext_vector_type(2))) float v2f;
typedef __attribute__((ext_vector_type(4))) float v4f;
typedef __attribute__((ext_vector_type(8))) float v8f;

#define DIM      1024
#define TILE     16
#define NTHREADS 256
#define NWAVES   8
#define KSLICE   (DIM / NWAVES) // 128 K per wave

__global__ __launch_bounds__(NTHREADS) void cross_compress_kernel(
    const float* __restrict__ v, const float* __restrict__ e,
    const float* __restrict__ w_vv, const float* __restrict__ w_ev,
    const float* __restrict__ w_ve, const float* __restrict__ w_ee,
    const float* __restrict__ bias_v, const float* __restrict__ bias_e,
    float* __restrict__ v_out, float* __restrict__ e_out)
{
    __shared__ float s_part[NWAVES][TILE][4]; // per-wave partial dots
    __shared__ float s_dot[TILE][4];          // reduced dots per row

    const int tid   = threadIdx.x;
    const int wave  = tid >> 5;
    const int lane  = tid & 31;
    const int rbase = blockIdx.x * TILE;

    // ---- Phase 1: dots via WMMA f32 16x16x4 ----
    // A layout (16x4 f32): lane m (m=lane&15) holds X[m][k0 + 2*(lane>>4) + j] in VGPR j.
    // B layout (4x16 f32): lane n (n=lane&15) holds W[k0 + 2*(lane>>4) + j][n] in VGPR j.
    // Columns: 0 = w_vv (pairs with A=e), 1 = w_ev (A=v), 2 = w_ve (A=e), 3 = w_ee (A=v).
    const int c     = lane & 15;
    const int khalf = lane >> 4;

    const float* arow_v = v + (size_t)(rbase + c) * DIM;
    const float* arow_e = e + (size_t)(rbase + c) * DIM;

    const float* wcol = w_vv;                 // lanes c>=4 read w_vv harmlessly, then zeroed
    if (c == 1) wcol = w_ev;
    else if (c == 2) wcol = w_ve;
    else if (c == 3) wcol = w_ee;
    const bool isE = (c == 0) || (c == 2);
    const bool isV = (c == 1) || (c == 3);

    v8f acc_e = {};
    v8f acc_v = {};

    const int kbeg = wave * KSLICE;
    #pragma unroll 4
    for (int k0 = kbeg; k0 < kbeg + KSLICE; k0 += 4) {
        const int kk = k0 + 2 * khalf;
        v2f ae = *(const v2f*)(arow_e + kk);
        v2f av = *(const v2f*)(arow_v + kk);
        const float w0 = wcol[kk];
        const float w1 = wcol[kk + 1];
        v2f be = { isE ? w0 : 0.0f, isE ? w1 : 0.0f };
        v2f bv = { isV ? w0 : 0.0f, isV ? w1 : 0.0f };
        // D = A*B + C ; two accumulators (disjoint output columns) to halve the dep chain
        acc_e = __builtin_amdgcn_wmma_f32_16x16x4_f32(false, ae, false, be,
                                                      (short)0, acc_e, false, false);
        acc_v = __builtin_amdgcn_wmma_f32_16x16x4_f32(false, av, false, bv,
                                                      (short)0, acc_v, false, false);
    }
    v8f acc = acc_e + acc_v;

    // D layout: lanes 0-15: VGPR j -> D[j][lane]; lanes 16-31: VGPR j -> D[8+j][lane-16].
    if (c < 4) {
        const int mbase = khalf * 8;
        #pragma unroll
        for (int j = 0; j < 8; ++j)
            s_part[wave][mbase + j][c] = acc[j];
    }
    __syncthreads();

    if (tid < TILE * 4) {
        const int m = tid >> 2, n = tid & 3;
        float s = 0.0f;
        #pragma unroll
        for (int w = 0; w < NWAVES; ++w) s += s_part[w][m][n];
        s_dot[m][n] = s;
    }
    __syncthreads();

    // ---- Phase 2: streaming broadcast-FMA (tile is L2-hot from phase 1) ----
    // v_out = v*s0 + e*s1 + bias_v ; e_out = v*s2 + e*s3 + bias_e
    #pragma unroll 4
    for (int i = tid; i < TILE * (DIM / 4); i += NTHREADS) {
        const int m  = i >> 8;                 // DIM/4 == 256
        const int c4 = i & (DIM / 4 - 1);
        const size_t off = (size_t)(rbase + m) * DIM + (size_t)c4 * 4;
        v4f vv  = *(const v4f*)(v + off);
        v4f ee  = *(const v4f*)(e + off);
        v4f bv4 = *(const v4f*)(bias_v + c4 * 4);
        v4f be4 = *(const v4f*)(bias_e + c4 * 4);
        const float s0 = s_dot[m][0];
        const float s1 = s_dot[m][1];
        const float s2 = s_dot[m][2];
        const float s3 = s_dot[m][3];
        v4f vo = vv * s0 + ee * s1 + bv4;
        v4f eo = vv * s2 + ee * s3 + be4;
        // write-once outputs: non-temporal so they don't evict streaming inputs from L2
        __builtin_nontemporal_store(vo, (v4f*)(v_out + off));
        __builtin_nontemporal_store(eo, (v4f*)(e_out + off));
    }
}

extern "C" void kernel_launch(void* const* d_in, const int* in_sizes, int n_in,
                              void* d_out, int out_size, void* d_ws, size_t ws_size,
                              hipStream_t stream) {
    const float* v    = (const float*)d_in[0];
    const float* e    = (const float*)d_in[1];
    const float* w_vv = (const float*)d_in[2];
    const float* w_ev = (const float*)d_in[3];
    const float* w_ve = (const float*)d_in[4];
    const float* w_ee = (const float*)d_in[5];
    const float* b_v  = (const float*)d_in[6];
    const float* b_e  = (const float*)d_in[7];

    const int batch = in_sizes[0] / DIM;      // 16384
    float* v_out = (float*)d_out;
    float* e_out = v_out + (size_t)batch * DIM;

    dim3 grid(batch / TILE);                  // 1024 tiles of 16 rows
    cross_compress_kernel<<<grid, NTHREADS, 0, stream>>>(
        v, e, w_vv, w_ev, w_ve, w_ee, b_v, b_e, v_out, e_out);
}